// AtmoSphericNeuralOperatorNet_48069273977528
// MI455X (gfx1250) — compile-verified
//
#include <hip/hip_runtime.h>
#include <math.h>

#define C_IN 32
#define H_IN 240
#define W_IN 480
#define H_OUT 481
#define W_OUT 960
#define C_OUT 73
#define K_BASIS 9
#define NNZ 96
#define HIDDEN 64
#define HW_IN (H_IN * W_IN)    /* 115200 */
#define HW_OUT (H_OUT * W_OUT) /* 461760 */
#define CK (C_IN * K_BASIS)    /* 288 */
#define MPAD 80                /* 73 output channels padded to 5 wmma M-tiles */
#define WT 32                  /* w-tile per workgroup */
#define YSTRIDE 40             /* 2*40 mod 64 = 16 -> stage-B half-wave LDS reads on disjoint banks */
#define NWAVES 16
#define MAIN_THREADS (NWAVES * 32) /* 512 */
#define KSTEPS (NNZ / 4)       /* 24 wmma steps over j for stage A */

typedef __attribute__((ext_vector_type(2))) float v2f;
typedef __attribute__((ext_vector_type(8))) float v8f;

__device__ __forceinline__ float gelu_tanh(float x) {
  float x3 = x * x * x;
  return 0.5f * x * (1.0f + tanhf(0.7978845608028654f * (x + 0.044715f * x3)));
}

// ---------------------------------------------------------------------------
// Kernel 0: transpose/pad weight (C_OUT,C_IN,K_BASIS) -> wT[ck=c*9+k][o(padded 80)]
// ---------------------------------------------------------------------------
__global__ __launch_bounds__(256) void asno_prep_weight(
    const float* __restrict__ weight, float* __restrict__ wT) {
  int t = blockIdx.x * 256 + threadIdx.x;
  if (t >= CK * MPAD) return;
  int o = t % MPAD;
  int ck = t / MPAD;
  int c = ck / K_BASIS, k = ck - c * K_BASIS;
  wT[t] = (o < C_OUT) ? weight[(o * C_IN + c) * K_BASIS + k] : 0.0f;
}

// ---------------------------------------------------------------------------
// Kernel 1: fused gelu -> 32x64 MLP -> gelu -> 64x32 MLP, one pixel per lane
// ---------------------------------------------------------------------------
__global__ __launch_bounds__(256) void asno_mlp(
    const float* __restrict__ x, const float* __restrict__ w1,
    const float* __restrict__ w2, float* __restrict__ hbuf) {
  __shared__ float w1s[HIDDEN * C_IN];
  __shared__ float w2s[C_IN * HIDDEN];
  for (int t = threadIdx.x; t < HIDDEN * C_IN; t += 256) {
    w1s[t] = w1[t];
    w2s[t] = w2[t];
  }
  __syncthreads();
  int p = blockIdx.x * 256 + threadIdx.x; /* HW_IN == 450*256 exactly */
  float xv[C_IN], h2[C_IN];
#pragma unroll
  for (int c = 0; c < C_IN; ++c) xv[c] = gelu_tanh(x[c * HW_IN + p]);
#pragma unroll
  for (int c = 0; c < C_IN; ++c) h2[c] = 0.0f;
  for (int d = 0; d < HIDDEN; ++d) {
    float s = 0.0f;
#pragma unroll
    for (int c = 0; c < C_IN; ++c) s = fmaf(w1s[d * C_IN + c], xv[c], s);
    float g = gelu_tanh(s);
#pragma unroll
    for (int c = 0; c < C_IN; ++c) h2[c] = fmaf(w2s[c * HIDDEN + d], g, h2[c]);
  }
#pragma unroll
  for (int c = 0; c < C_IN; ++c) hbuf[c * HW_IN + p] = h2[c];
}

// ---------------------------------------------------------------------------
// Kernel 2: bilinear S2 upsample (240x480 -> 481x960) fused with quad_w scale
// ---------------------------------------------------------------------------
__global__ __launch_bounds__(256) void asno_upsample(
    const float* __restrict__ hbuf, const float* __restrict__ quad_w,
    float* __restrict__ xq) {
  int p = blockIdx.x * 256 + threadIdx.x;
  if (p >= HW_OUT) return;
  int ho = p / W_OUT;
  int wo = p - ho * W_OUT;
  float pos = (ho + 0.5f) * ((float)H_IN / (float)H_OUT) - 0.5f;
  int i0 = (int)floorf(pos);
  i0 = i0 < 0 ? 0 : (i0 > H_IN - 2 ? H_IN - 2 : i0);
  float fl = pos - (float)i0;
  fl = fl < 0.f ? 0.f : (fl > 1.f ? 1.f : fl);
  int j0 = wo >> 1; /* floor(wo*0.5) < 480 always */
  float fw = (wo & 1) ? 0.5f : 0.0f;
  int j1 = j0 + 1;
  if (j1 >= W_IN) j1 -= W_IN;
  float qw = quad_w[ho];
  float c00 = (1.f - fl) * (1.f - fw) * qw;
  float c01 = (1.f - fl) * fw * qw;
  float c10 = fl * (1.f - fw) * qw;
  float c11 = fl * fw * qw;
  const float* r0 = hbuf + i0 * W_IN;
  const float* r1 = r0 + W_IN;
  for (int c = 0; c < C_IN; ++c) {
    int off = c * HW_IN;
    xq[c * HW_OUT + p] = c00 * r0[off + j0] + c01 * r0[off + j1] +
                         c10 * r1[off + j0] + c11 * r1[off + j1];
  }
}

// ---------------------------------------------------------------------------
// Kernel 3: per (row h, 32-col tile). Both stages on the f32 matrix pipe:
//   Stage A (WMMA): Y[k, n=(c,w)] = sum_j Vpad[k,j] * xq[c, hi[j], (dwi[j]+w)%960]
//     - B-fragment = 2 gathered global loads per lane (the gathers ARE the operand)
//     - A-fragments (conv_vals row, 9->16 zero-pad) preloaded once per wave (48 VGPRs)
//   Stage B (WMMA): out[o, w] = sum_ck wT[ck][o] * y_lds[ck][w]
// ---------------------------------------------------------------------------
__global__ __launch_bounds__(MAIN_THREADS) void asno_spherical_conv_wmma(
    const float* __restrict__ xq, const float* __restrict__ wT,
    const float* __restrict__ conv_vals, const int* __restrict__ conv_hi,
    const int* __restrict__ conv_dwi, float* __restrict__ out) {
  __shared__ float y_lds[CK * YSTRIDE];  /* 46080 B : stage-A out / stage-B in */
  __shared__ int cidx_lds[NNZ * WT];     /* 12288 B : precomputed gather indices */
  __shared__ float vpad_lds[NNZ * 16];   /*  6144 B : conv_vals row, j-major, k padded to 16 */

  const int h = blockIdx.y;
  const int w0 = blockIdx.x * WT;
  const int tid = threadIdx.x;
  const int wave = tid >> 5; /* wave32 */
  const int lane = tid & 31;
  const int half = lane >> 4;
  const int r = lane & 15;

  /* Build gather-index table: cidx[j][w] = hi[j]*960 + (dwi[j]+w0+w) mod 960 */
  for (int t = tid; t < NNZ * WT; t += MAIN_THREADS) {
    int j = t >> 5;
    int w = t & (WT - 1);
    int col = conv_dwi[h * NNZ + j] + w0 + w;
    col -= (col >= W_OUT) ? W_OUT : 0;
    cidx_lds[t] = conv_hi[h * NNZ + j] * W_OUT + col;
  }
  /* Build padded A-matrix for stage A: vpad[j][m] = (m<9) ? conv_vals[m,h,j] : 0 */
  for (int t = tid; t < NNZ * 16; t += MAIN_THREADS) {
    int j = t >> 4;
    int m = t & 15;
    vpad_lds[t] = (m < K_BASIS) ? conv_vals[(m * H_OUT + h) * NNZ + j] : 0.0f;
  }
  __syncthreads();

  /* Preload all 24 stage-A A-fragments (wave-invariant across jobs).
     Half-waves read j-rows 32 floats apart -> disjoint LDS banks. */
  v2f afrag[KSTEPS];
#pragma unroll
  for (int s = 0; s < KSTEPS; ++s) {
    int j0 = s * 4 + 2 * half;
    afrag[s].x = vpad_lds[j0 * 16 + r];
    afrag[s].y = vpad_lds[(j0 + 1) * 16 + r];
  }

  /* Stage A: 64 n-tiles (c 0..31 x two 16-col segments), 4 per wave. */
  for (int jb = 0; jb < 64; jb += NWAVES) {
    const int job = jb + wave;
    const int c = job >> 1;
    const int wseg = job & 1;
    const int n = wseg * 16 + r;
    const float* pc = xq + c * HW_OUT;
    v8f acc = {};
#pragma unroll
    for (int s = 0; s < KSTEPS; ++s) {
      int j0 = s * 4 + 2 * half;
      v2f b;
      b.x = pc[cidx_lds[j0 * WT + n]];
      b.y = pc[cidx_lds[(j0 + 1) * WT + n]];
      acc = __builtin_amdgcn_wmma_f32_16x16x4_f32(false, afrag[s], false, b,
                                                  (short)0, acc, false, false);
    }
    /* D: VGPR r8 holds M=r8+8*half (k index); keep the 9 valid rows. */
#pragma unroll
    for (int r8 = 0; r8 < 8; ++r8) {
      int k = r8 + 8 * half;
      if (k < K_BASIS) y_lds[(c * K_BASIS + k) * YSTRIDE + n] = acc[r8];
    }
  }
  __syncthreads();

  /* Stage B: 10 output tiles (5 o-tiles x 2 w-segments) on waves 0..9.
     Wave-uniform branch: EXEC stays all-ones inside active waves. */
  if (wave < 10) {
    const int mt = wave >> 1; /* 0..4 */
    const int nt = wave & 1;  /* 0..1 */
    v8f acc = {};
    const float* wTa = wT + mt * 16 + r;
    const float* yb = y_lds + nt * 16 + r;
    for (int kk = 0; kk < CK; kk += 4) {
      v2f a, b;
      a.x = wTa[(kk + 2 * half) * MPAD];
      a.y = wTa[(kk + 1 + 2 * half) * MPAD];
      b.x = yb[(kk + 2 * half) * YSTRIDE];
      b.y = yb[(kk + 1 + 2 * half) * YSTRIDE];
      acc = __builtin_amdgcn_wmma_f32_16x16x4_f32(false, a, false, b, (short)0,
                                                  acc, false, false);
    }
    const int wcol = w0 + nt * 16 + r;
#pragma unroll
    for (int r8 = 0; r8 < 8; ++r8) {
      int o = mt * 16 + half * 8 + r8;
      if (o < C_OUT) out[o * HW_OUT + h * W_OUT + wcol] = acc[r8];
    }
  }
}

// ---------------------------------------------------------------------------
extern "C" void kernel_launch(void* const* d_in, const int* in_sizes, int n_in,
                              void* d_out, int out_size, void* d_ws,
                              size_t ws_size, hipStream_t stream) {
  (void)in_sizes; (void)n_in; (void)out_size; (void)ws_size;
  const float* x = (const float*)d_in[0];
  const float* w1 = (const float*)d_in[1];
  const float* w2 = (const float*)d_in[2];
  const float* cvals = (const float*)d_in[3];
  const float* qw = (const float*)d_in[4];
  const float* weight = (const float*)d_in[5];
  const int* chi = (const int*)d_in[6];
  const int* cdwi = (const int*)d_in[7];
  float* out = (float*)d_out;

  float* hbuf = (float*)d_ws;              /* 32*240*480 f32  (14.7 MB) */
  float* xq = hbuf + (size_t)C_IN * HW_IN; /* 32*481*960 f32  (59.1 MB) */
  float* wT = xq + (size_t)C_IN * HW_OUT;  /* 288*80 f32      (92 KB)   */

  asno_prep_weight<<<(CK * MPAD + 255) / 256, 256, 0, stream>>>(weight, wT);
  asno_mlp<<<HW_IN / 256, 256, 0, stream>>>(x, w1, w2, hbuf);
  asno_upsample<<<(HW_OUT + 255) / 256, 256, 0, stream>>>(hbuf, qw, xq);
  dim3 grid(W_OUT / WT, H_OUT);
  asno_spherical_conv_wmma<<<grid, MAIN_THREADS, 0, stream>>>(xq, wT, cvals,
                                                              chi, cdwi, out);
}